// OBJ_8409545965931
// MI455X (gfx1250) — compile-verified
//
#include <hip/hip_runtime.h>
#include <math.h>

// ---------------------------------------------------------------------------
// Problem constants (from reference)
// ---------------------------------------------------------------------------
#define NN      6144
#define NFEAT   500
#define HID     128
#define NCLASS  10
#define KLAYERS 3
#define EPSN    1e-12f
#define NJT     (NN / 16)      // 384 j-tiles

typedef float v2f __attribute__((ext_vector_type(2)));
typedef float v8f __attribute__((ext_vector_type(8)));
typedef unsigned int u32x4 __attribute__((ext_vector_type(4)));
typedef int i32x4 __attribute__((ext_vector_type(4)));
typedef int i32x8 __attribute__((ext_vector_type(8)));

// V_WMMA_F32_16X16X4_F32 : D[16x16,f32] = A[16x4,f32] * B[4x16,f32] + C
static __device__ __forceinline__ v8f wmma4(v2f a, v2f b, v8f c) {
    return __builtin_amdgcn_wmma_f32_16x16x4_f32(false, a, false, b,
                                                 (short)0, c, false, false);
}

// ---------------------------------------------------------------------------
// TDM: 1-D contiguous tile load (nelem 4-byte words) global -> LDS.
// D# per CDNA5 ISA: group0 = {flags, lds_addr, global_addr,
// type=2}; group1 = {data_size=4B, tensor_dim0, tensor_dim1=1, tile_dim0,
// tile_dim1=0 (1-D), stride0}; groups 2/3 unused (<=2D tensor).
// Issued per-wave (EXEC ignored); tracked with TENSORcnt.
// This toolchain exposes the 6-arg builtin: (g0, g1, g2, g3, g4, cpol).
// ---------------------------------------------------------------------------
static __device__ __forceinline__ void tdm_load_1d(const float* gptr,
                                                   unsigned lds_off,
                                                   int nelem) {
    unsigned long long ga = (unsigned long long)(uintptr_t)gptr;
    u32x4 g0;
    g0[0] = 1u;                                   // count=1, user load
    g0[1] = lds_off;                              // lds_addr [63:32]
    g0[2] = (unsigned)(ga & 0xFFFFFFFFu);         // global_addr lo
    g0[3] = (unsigned)((ga >> 32) & 0x1FFFFFFu)   // global_addr hi (57-bit)
          | (2u << 30);                           // type=2 ("image")
    i32x8 g1;
    g1[0] = (2 << 16);                            // data_size=2 -> 4 bytes
    g1[1] = (nelem & 0xFFFF) << 16;               // tensor_dim0[15:0]
    g1[2] = ((nelem >> 16) & 0xFFFF) | (1 << 16); // tensor_dim0 hi, tensor_dim1=1
    g1[3] = (nelem & 0xFFFF) << 16;               // tile_dim0
    g1[4] = 0;                                    // tile_dim1=0, tile_dim2=0
    g1[5] = nelem;                                // tensor_dim0_stride lo
    g1[6] = 0;
    g1[7] = 0;
    i32x4 z4 = {0, 0, 0, 0};
    i32x8 z8 = {0, 0, 0, 0, 0, 0, 0, 0};
    __builtin_amdgcn_tensor_load_to_lds(g0, g1, z4, z4, z8, 0);
}

static __device__ __forceinline__ unsigned lds_offset_of(const void* p) {
    return (unsigned)(uintptr_t)p;   // low 32 bits of generic ptr == LDS offset
}

// ---------------------------------------------------------------------------
// Generic C[N x 128] = A[N x KDIM] * W^T  (+ bias), W is [128 x KDIM]
// One wave computes one 16x16 tile.  4 waves / block.
// ---------------------------------------------------------------------------
template <int KDIM, bool BIAS>
__global__ void gemm_qwT(const float* __restrict__ A,
                         const float* __restrict__ W,
                         const float* __restrict__ bias,
                         float* __restrict__ C) {
    const int wave = threadIdx.x >> 5;
    const int lane = threadIdx.x & 31;
    const bool hi  = lane >= 16;
    const int  c   = lane & 15;
    const int  koff = hi ? 2 : 0;

    const int gt = blockIdx.x * 4 + wave;   // 3072 tiles total
    const int m0 = (gt >> 3) * 16;          // row tile
    const int n0 = (gt & 7) * 16;           // col tile

    v8f acc = {0.f, 0.f, 0.f, 0.f, 0.f, 0.f, 0.f, 0.f};

    const float* arow = A + (m0 + c) * KDIM;
    const float* wrow = W + (n0 + c) * KDIM;

#pragma unroll 4
    for (int k = 0; k < KDIM; k += 4) {
        v2f a = *(const v2f*)(arow + k + koff);
        v2f b = *(const v2f*)(wrow + k + koff);
        acc = wmma4(a, b, acc);
    }

    if (BIAS) {
        const float bv = bias[n0 + c];
#pragma unroll
        for (int r = 0; r < 8; ++r) acc[r] += bv;
    }

    const int rbase = m0 + (hi ? 8 : 0);
#pragma unroll
    for (int r = 0; r < 8; ++r)
        C[(rbase + r) * HID + n0 + c] = acc[r];
}

// ---------------------------------------------------------------------------
// Fused gate + aggregation with TDM double-buffered j-tile staging:
//   h_i = h1_i + sum_j [ g_ij*h3_j + (g_ij - a_ij)*h4_j ],
//   g_ij = adj_ij * sigmoid(-(h3_i . h4_j))
// 4 waves / block, 16 rows per wave (64 rows / block).
// ---------------------------------------------------------------------------
__global__ void fused_gate_agg(const float* __restrict__ H1,
                               const float* __restrict__ H3,
                               const float* __restrict__ H4,
                               const float* __restrict__ adj,
                               float* __restrict__ Hout) {
    __shared__ float sbuf[2][2][16 * HID];   // [buffer][h3|h4][tile] : 32 KB
    __shared__ float sG [4][16 * 16];        // gate tile  g      per wave
    __shared__ float sGM[4][16 * 16];        // gate tile  g - a  per wave

    const int wave = threadIdx.x >> 5;
    const int lane = threadIdx.x & 31;
    const bool hi  = lane >= 16;
    const int  c   = lane & 15;
    const int  koff = hi ? 2 : 0;
    const int  row0 = blockIdx.x * 64 + wave * 16;

    // Cache this wave's A-fragments of h3 rows (16 x 128) : 32 v2f = 64 VGPRs
    v2f aF[32];
    {
        const float* r = H3 + (row0 + c) * HID;
#pragma unroll
        for (int kk = 0; kk < 32; ++kk)
            aF[kk] = *(const v2f*)(r + 4 * kk + koff);
    }

    // Accumulators init from H1 (self-loop term), C/D layout
    v8f acc[8];
    {
        const int rbase = row0 + (hi ? 8 : 0);
#pragma unroll
        for (int t = 0; t < 8; ++t)
#pragma unroll
            for (int r = 0; r < 8; ++r)
                acc[t][r] = H1[(rbase + r) * HID + t * 16 + c];
    }

    // Prologue: wave 0 issues TDM load of j-tile 0 into buffer 0
    if (threadIdx.x < 32) {
        tdm_load_1d(H3, lds_offset_of(&sbuf[0][0][0]), 16 * HID);
        tdm_load_1d(H4, lds_offset_of(&sbuf[0][1][0]), 16 * HID);
    }

    for (int jb = 0; jb < NJT; ++jb) {
        const int j0  = jb * 16;
        const int cur = jb & 1;

        __syncthreads();   // previous iteration done reading buffer cur^1
        if (threadIdx.x < 32) {
            if (jb + 1 < NJT) {
                // prefetch next j-tile into the other buffer (overlaps compute)
                tdm_load_1d(H3 + (j0 + 16) * HID,
                            lds_offset_of(&sbuf[cur ^ 1][0][0]), 16 * HID);
                tdm_load_1d(H4 + (j0 + 16) * HID,
                            lds_offset_of(&sbuf[cur ^ 1][1][0]), 16 * HID);
                // in-order completion: <=2 outstanding => current tile landed
                __builtin_amdgcn_s_wait_tensorcnt(2);
            } else {
                __builtin_amdgcn_s_wait_tensorcnt(0);
            }
        }
        __syncthreads();   // current buffer visible to all waves

        const float* sh3 = sbuf[cur][0];
        const float* sh4 = sbuf[cur][1];

        // ---- Gate matmul: pre = h3_rows . h4_j^T  (16x16, K=128) ----
        v8f sacc = {0.f, 0.f, 0.f, 0.f, 0.f, 0.f, 0.f, 0.f};
#pragma unroll 8
        for (int kk = 0; kk < 32; ++kk) {
            v2f b = *(const v2f*)(sh4 + c * HID + 4 * kk + koff);
            sacc = wmma4(aF[kk], b, sacc);
        }

        // ---- sigmoid(-pre) * adj, build g and g-a in LDS ----
        const int rbase = row0 + (hi ? 8 : 0);
        if (j0 + 16 < NN)
            __builtin_prefetch(adj + (size_t)(rbase)*NN + j0 + 16 + c, 0, 0);
#pragma unroll
        for (int r = 0; r < 8; ++r) {
            const float a = adj[(rbase + r) * NN + j0 + c];
            const float s = 1.0f / (1.0f + expf(sacc[r]));  // sigmoid(-x)
            const float g = a * s;
            const int m = r + (hi ? 8 : 0);
            sG [wave][m * 16 + c] = g;
            sGM[wave][m * 16 + c] = g - a;
        }
        __syncthreads();

        // ---- Aggregation: acc += g @ h3_j + (g-a) @ h4_j  (K=16) ----
        const float* sg  = sG[wave];
        const float* sgm = sGM[wave];
#pragma unroll
        for (int kk = 0; kk < 4; ++kk) {
            const int kb = 4 * kk + koff;
            v2f ag = *(const v2f*)(sg  + c * 16 + kb);
            v2f am = *(const v2f*)(sgm + c * 16 + kb);
#pragma unroll
            for (int t = 0; t < 8; ++t) {
                const int n = t * 16 + c;
                v2f b3 = {sh3[kb * HID + n], sh3[(kb + 1) * HID + n]};
                v2f b4 = {sh4[kb * HID + n], sh4[(kb + 1) * HID + n]};
                acc[t] = wmma4(ag, b3, acc[t]);
                acc[t] = wmma4(am, b4, acc[t]);
            }
        }
    }

    // Store new h
    const int rbase = row0 + (hi ? 8 : 0);
#pragma unroll
    for (int t = 0; t < 8; ++t)
#pragma unroll
        for (int r = 0; r < 8; ++r)
            Hout[(rbase + r) * HID + t * 16 + c] = acc[t][r];
}

// ---------------------------------------------------------------------------
// Q[row] = H[row] / max(||H[row]||_2, eps) ; one wave per row
// ---------------------------------------------------------------------------
__global__ void normalize_rows(const float* __restrict__ H,
                               float* __restrict__ Q) {
    const int wave = threadIdx.x >> 5;
    const int lane = threadIdx.x & 31;
    const int row  = blockIdx.x * 8 + wave;

    float4 v = ((const float4*)(H + row * HID))[lane];
    float ss = v.x * v.x + v.y * v.y + v.z * v.z + v.w * v.w;
#pragma unroll
    for (int m = 16; m > 0; m >>= 1) ss += __shfl_xor(ss, m, 32);
    const float inv = 1.0f / fmaxf(sqrtf(ss), EPSN);
    float4 o = {v.x * inv, v.y * inv, v.z * inv, v.w * inv};
    ((float4*)(Q + row * HID))[lane] = o;
}

// ---------------------------------------------------------------------------
// logits = Q @ W_out^T + b_out ; out = [log_softmax(logits) ; logits]
// ---------------------------------------------------------------------------
__global__ void head_logsoftmax(const float* __restrict__ Q,
                                const float* __restrict__ Wout,
                                const float* __restrict__ bout,
                                float* __restrict__ out) {
    const int row = blockIdx.x * blockDim.x + threadIdx.x;
    const float* q = Q + row * HID;

    float lg[NCLASS];
#pragma unroll
    for (int cc = 0; cc < NCLASS; ++cc) {
        float d = bout[cc];
        const float* w = Wout + cc * HID;
#pragma unroll 8
        for (int k = 0; k < HID; ++k) d += q[k] * w[k];
        lg[cc] = d;
    }
    float mx = lg[0];
#pragma unroll
    for (int cc = 1; cc < NCLASS; ++cc) mx = fmaxf(mx, lg[cc]);
    float se = 0.f;
#pragma unroll
    for (int cc = 0; cc < NCLASS; ++cc) se += expf(lg[cc] - mx);
    const float lse = logf(se);
#pragma unroll
    for (int cc = 0; cc < NCLASS; ++cc) {
        out[row * NCLASS + cc]               = lg[cc] - mx - lse;  // log_softmax
        out[NN * NCLASS + row * NCLASS + cc] = lg[cc];             // logits
    }
}

// ---------------------------------------------------------------------------
extern "C" void kernel_launch(void* const* d_in, const int* in_sizes, int n_in,
                              void* d_out, int out_size, void* d_ws, size_t ws_size,
                              hipStream_t stream) {
    const float* x     = (const float*)d_in[0];
    const float* adj   = (const float*)d_in[1];
    const float* W_in  = (const float*)d_in[2];
    const float* b_in  = (const float*)d_in[3];
    const float* W1    = (const float*)d_in[4];
    const float* W2    = (const float*)d_in[5];
    const float* Wf    = (const float*)d_in[6];
    const float* W_out = (const float*)d_in[7];
    const float* b_out = (const float*)d_in[8];

    float* ws = (float*)d_ws;
    float* Q  = ws;
    float* H1 = Q  + NN * HID;
    float* H3 = H1 + NN * HID;
    float* H4 = H3 + NN * HID;
    float* Hn = H4 + NN * HID;

    // Q = x @ W_in^T + b_in
    gemm_qwT<NFEAT, true><<<768, 128, 0, stream>>>(x, W_in, b_in, Q);

    for (int i = 0; i < KLAYERS; ++i) {
        const int off = i * HID * HID;
        gemm_qwT<HID, false><<<768, 128, 0, stream>>>(Q, W1 + off, nullptr, H3);
        gemm_qwT<HID, false><<<768, 128, 0, stream>>>(Q, W2 + off, nullptr, H4);
        gemm_qwT<HID, false><<<768, 128, 0, stream>>>(Q, Wf + off, nullptr, H1);
        fused_gate_agg<<<96, 128, 0, stream>>>(H1, H3, H4, adj, Hn);
        normalize_rows<<<768, 256, 0, stream>>>(Hn, Q);
    }

    head_logsoftmax<<<24, 256, 0, stream>>>(Q, W_out, b_out, (float*)d_out);
}